// LightGCN_89472758710578
// MI455X (gfx1250) — compile-verified
//
#include <hip/hip_runtime.h>
#include <hip/hip_bf16.h>
#include <stdint.h>

#define DIM 64
#define TILE 64   // edges staged per wave per iteration

typedef float v4f __attribute__((ext_vector_type(4)));   // native vector for NT builtins

// ---------------------------------------------------------------------------
// gfx1250 async global->LDS staging (CDNA5 path, ASYNCcnt-tracked).
// Edge index streams are single-use: mark non-temporal so they don't evict
// the L2-resident gather/atomic working sets (h_user/rst/bsrc).
// ---------------------------------------------------------------------------
__device__ __forceinline__ void async_b64_to_lds(unsigned lds_addr, const void* gaddr) {
    asm volatile("global_load_async_to_lds_b64 %0, %1, off th:TH_LOAD_NT"
                 :: "v"(lds_addr), "v"((unsigned long long)(uintptr_t)gaddr)
                 : "memory");
}
__device__ __forceinline__ void async_b32_to_lds(unsigned lds_addr, const void* gaddr) {
    asm volatile("global_load_async_to_lds_b32 %0, %1, off th:TH_LOAD_NT"
                 :: "v"(lds_addr), "v"((unsigned long long)(uintptr_t)gaddr)
                 : "memory");
}
__device__ __forceinline__ void wait_async_zero() {
    asm volatile("s_wait_asynccnt 0" ::: "memory");
}
// Low 32 bits of a flat LDS pointer == LDS byte offset.
__device__ __forceinline__ unsigned lds_off(const void* p) {
    return (unsigned)(uintptr_t)p;
}
// Stage up to TILE edge indices for this wave; lane pulls 2 consecutive ints.
__device__ __forceinline__ void stage_pair(unsigned ls, unsigned ld,
                                           const int* esrc, const int* edst,
                                           long base, int lane, int cnt) {
    int o = lane * 2;
    if (o + 1 < cnt) {
        async_b64_to_lds(ls, esrc + base + o);
        async_b64_to_lds(ld, edst + base + o);
    } else if (o < cnt) {      // odd tail element
        async_b32_to_lds(ls, esrc + base + o);
        async_b32_to_lds(ld, edst + base + o);
    }
}

// Native f32 atomic add (global_atomic_add_f32; resolves in 192MB L2).
__device__ __forceinline__ void fadd(float* p, float v) { unsafeAtomicAdd(p, v); }

// ---------------------------------------------------------------------------
// Zero-fill (float4 main + tail). Regular TH: these lines are re-read/RMW'd
// by the scatter atomics right after, so keep them cacheable.
// ---------------------------------------------------------------------------
__global__ void __launch_bounds__(256) zero_f32(float* __restrict__ p, long n) {
    long i4 = (long)blockIdx.x * blockDim.x + threadIdx.x;
    long n4 = n >> 2;
    long stride = (long)gridDim.x * blockDim.x;
    v4f z = {0.f, 0.f, 0.f, 0.f};
    for (long i = i4; i < n4; i += stride) ((v4f*)p)[i] = z;
    for (long i = (n4 << 2) + i4; i < n; i += stride) p[i] = 0.f;
}

// ---------------------------------------------------------------------------
// Forward: per wave, async-stage TILE (src,dst) pairs into LDS, bump degrees
// (2 edges per lane), then scatter h_user rows into rst. 16 lanes cover one
// 64-float row via float4 -> 2 edges per inner iteration.
// ---------------------------------------------------------------------------
__global__ void __launch_bounds__(256)
fwd_scatter(const float* __restrict__ h_user,
            const int* __restrict__ esrc, const int* __restrict__ edst,
            float* __restrict__ rst, float* __restrict__ deg_dst,
            float* __restrict__ deg_src, int n_edge) {
    __shared__ int s_src[8 * TILE];
    __shared__ int s_dst[8 * TILE];
    const int lane  = threadIdx.x & 31;
    const int widx  = threadIdx.x >> 5;
    const int wbase = widx * TILE;
    const unsigned ls = lds_off(&s_src[wbase + lane * 2]);
    const unsigned ld = lds_off(&s_dst[wbase + lane * 2]);
    const int half = lane >> 4;        // which of 2 edges this half-wave handles
    const int sub  = lane & 15;        // float4 slot within the row

    long gw = (long)blockIdx.x * 8 + widx;
    long nw = (long)gridDim.x * 8;
    for (long base = gw * TILE; base < n_edge; base += nw * TILE) {
        int cnt = (n_edge - base < TILE) ? (int)(n_edge - base) : TILE;
        stage_pair(ls, ld, esrc, edst, base, lane, cnt);
        wait_async_zero();
        // degree accumulation: 2 edges per lane
        if (lane < cnt) {
            fadd(&deg_src[s_src[wbase + lane]], 1.0f);
            fadd(&deg_dst[s_dst[wbase + lane]], 1.0f);
        }
        if (lane + 32 < cnt) {
            fadd(&deg_src[s_src[wbase + 32 + lane]], 1.0f);
            fadd(&deg_dst[s_dst[wbase + 32 + lane]], 1.0f);
        }
        for (int i = 0; i < cnt; i += 2) {
            int j = i + half;
            if (j < cnt) {
                int src = s_src[wbase + j];
                int dst = s_dst[wbase + j];
                const v4f v = *(const v4f*)(h_user + (size_t)src * DIM + sub * 4);
                float* r = rst + (size_t)dst * DIM + sub * 4;
                fadd(r + 0, v.x); fadd(r + 1, v.y);
                fadd(r + 2, v.z); fadd(r + 3, v.w);
            }
        }
    }
}

// ---------------------------------------------------------------------------
// Backward: gather normalized rst[dst] (25.6 MB, L2-hot) and scatter into
// bsrc[src] (128 MB, fits L2 -> atomics resolve on-chip).
// ---------------------------------------------------------------------------
__global__ void __launch_bounds__(256)
bwd_scatter(const float* __restrict__ rst,
            const int* __restrict__ esrc, const int* __restrict__ edst,
            float* __restrict__ bsrc, int n_edge) {
    __shared__ int s_src[8 * TILE];
    __shared__ int s_dst[8 * TILE];
    const int lane  = threadIdx.x & 31;
    const int widx  = threadIdx.x >> 5;
    const int wbase = widx * TILE;
    const unsigned ls = lds_off(&s_src[wbase + lane * 2]);
    const unsigned ld = lds_off(&s_dst[wbase + lane * 2]);
    const int half = lane >> 4;
    const int sub  = lane & 15;

    long gw = (long)blockIdx.x * 8 + widx;
    long nw = (long)gridDim.x * 8;
    for (long base = gw * TILE; base < n_edge; base += nw * TILE) {
        int cnt = (n_edge - base < TILE) ? (int)(n_edge - base) : TILE;
        stage_pair(ls, ld, esrc, edst, base, lane, cnt);
        wait_async_zero();
        for (int i = 0; i < cnt; i += 2) {
            int j = i + half;
            if (j < cnt) {
                int src = s_src[wbase + j];
                int dst = s_dst[wbase + j];
                const v4f v = *(const v4f*)(rst + (size_t)dst * DIM + sub * 4);
                float* r = bsrc + (size_t)src * DIM + sub * 4;
                fadd(r + 0, v.x); fadd(r + 1, v.y);
                fadd(r + 2, v.z); fadd(r + 3, v.w);
            }
        }
    }
}

// ---------------------------------------------------------------------------
// Row-wise scale by 1/max(deg,1) (covers deg^-1 for both passes; NORM_2=-1).
// Last-touch traffic: non-temporal vector load/store so it doesn't evict the
// scatter working sets from L2.
// ---------------------------------------------------------------------------
__global__ void __launch_bounds__(256)
norm_rows(float* __restrict__ data, const float* __restrict__ deg, long n_elem) {
    long i4 = (long)blockIdx.x * blockDim.x + threadIdx.x;
    long n4 = n_elem >> 2;               // DIM==64 -> always divisible by 4
    long stride = (long)gridDim.x * blockDim.x;
    v4f* p = (v4f*)data;
    for (long i = i4; i < n4; i += stride) {
        float d = deg[i >> 4];           // 16 float4 per row
        float s = 1.0f / fmaxf(d, 1.0f);
        v4f v = __builtin_nontemporal_load(&p[i]);
        v *= s;
        __builtin_nontemporal_store(v, &p[i]);
    }
}

extern "C" void kernel_launch(void* const* d_in, const int* in_sizes, int n_in,
                              void* d_out, int out_size, void* d_ws, size_t ws_size,
                              hipStream_t stream) {
    const float* h_user = (const float*)d_in[0];
    // d_in[1] = h_group: unused (ALPHA == 0)
    const int* esrc = (const int*)d_in[2];
    const int* edst = (const int*)d_in[3];

    const long n_user  = in_sizes[0] / DIM;   // 500,000
    const long n_group = in_sizes[1] / DIM;   // 100,000
    const int  n_edge  = in_sizes[2];         // 5,000,000

    float* out  = (float*)d_out;
    float* bsrc = out;                               // [n_user,  64]
    float* rst  = out + (size_t)n_user * DIM;        // [n_group, 64]
    float* deg_dst = (float*)d_ws;                   // n_group floats
    float* deg_src = deg_dst + n_group;              // n_user floats

    const int TB = 256;
    // 1) zero degree scratch + outputs (harness poisons them)
    zero_f32<<<2048, TB, 0, stream>>>(deg_dst, n_group + n_user);
    zero_f32<<<4096, TB, 0, stream>>>(out, (long)out_size);

    // 2) forward scatter + degree counts
    fwd_scatter<<<4096, TB, 0, stream>>>(h_user, esrc, edst, rst,
                                         deg_dst, deg_src, n_edge);

    // 3) rst *= 1/max(deg_dst,1)
    norm_rows<<<2048, TB, 0, stream>>>(rst, deg_dst, n_group * DIM);

    // 4) backward scatter
    bwd_scatter<<<4096, TB, 0, stream>>>(rst, esrc, edst, bsrc, n_edge);

    // 5) bsrc *= max(deg_src,1)^-1
    norm_rows<<<4096, TB, 0, stream>>>(bsrc, deg_src, n_user * DIM);
}